// ExpertLoRA_57750130262030
// MI455X (gfx1250) — compile-verified
//
#include <hip/hip_runtime.h>
#include <stdint.h>

#define E_ 16
#define H_ 1024
#define F_ 1024
#define D_ 2048
#define R_ 16
#define T_ 1024
#define TOPK_ 2
#define NPAIR_ (T_ * TOPK_)
#define CAPP_ (NPAIR_ + 32)
#define LIMIT_ 7.0f
#define ACTA_ 1.702f
#define SCALING_ (16.0f / (float)R_)

typedef __attribute__((ext_vector_type(2))) float v2f;
typedef __attribute__((ext_vector_type(8))) float v8f;

// ---------------------------------------------------------------------------
// CDNA5 async global->LDS copy (ASYNCcnt-tracked, no VGPR round trip).
// ---------------------------------------------------------------------------
__device__ __forceinline__ void async_copy_b128(unsigned lds_off, uint64_t gaddr)
{
    asm volatile("global_load_async_to_lds_b128 %0, %1, off"
                 :: "v"(lds_off), "v"(gaddr)
                 : "memory");
}
__device__ __forceinline__ void async_wait0()
{
    asm volatile("s_wait_asynccnt 0x0" ::: "memory");
}
__device__ __forceinline__ void async_wait1()
{
    asm volatile("s_wait_asynccnt 0x1" ::: "memory");
}

// ---------------------------------------------------------------------------
// Kernel 1: routing — build compacted per-expert pair lists.
// ---------------------------------------------------------------------------
__global__ __launch_bounds__(1024)
void moe_route(const int* __restrict__ ridx, const float* __restrict__ rw,
               int* __restrict__ offs, int* __restrict__ ptok,
               int* __restrict__ pexp, float* __restrict__ pw)
{
    __shared__ int scnt[E_], sbase[E_];
    int t = threadIdx.x;
    if (t < E_) scnt[t] = 0;
    __syncthreads();
    int loc[TOPK_], ee[TOPK_];
#pragma unroll
    for (int k = 0; k < TOPK_; ++k) {
        int e = ridx[t * TOPK_ + k];
        ee[k] = e;
        loc[k] = ((unsigned)e < (unsigned)E_) ? atomicAdd(&scnt[e], 1) : -1;
    }
    __syncthreads();
    if (t == 0) {
        int acc = 0;
        for (int e = 0; e < E_; ++e) { offs[e] = acc; sbase[e] = acc; acc += scnt[e]; }
        offs[E_] = acc;
    }
    __syncthreads();
#pragma unroll
    for (int k = 0; k < TOPK_; ++k) {
        if (loc[k] >= 0) {
            int p = sbase[ee[k]] + loc[k];
            ptok[p] = t;
            pexp[p] = ee[k];
            pw[p]   = rw[t * TOPK_ + k];
        }
    }
}

// ---------------------------------------------------------------------------
// Kernel 2: r_gu[p, :] = x[token(p), :] @ lora_gate_up_A[expert(p)]   (R=16)
// ---------------------------------------------------------------------------
__global__ __launch_bounds__(128)
void lora_r_gu(const float* __restrict__ x, const float* __restrict__ A,
               const int* __restrict__ offs, const int* __restrict__ ptok,
               const int* __restrict__ pexp, float* __restrict__ rout)
{
    int p = blockIdx.x;
    if (p >= offs[E_]) return;
    int tok = ptok[p], e = pexp[p];
    const float* xr = x + (size_t)tok * H_;
    const float* Ae = A + (size_t)e * H_ * R_;
    int r = threadIdx.x >> 3, l = threadIdx.x & 7;
    float acc = 0.f;
    for (int h = l; h < H_; h += 8) acc += xr[h] * Ae[(size_t)h * R_ + r];
    acc += __shfl_xor(acc, 1);
    acc += __shfl_xor(acc, 2);
    acc += __shfl_xor(acc, 4);
    if (l == 0) rout[(size_t)p * R_ + r] = acc;
}

// ---------------------------------------------------------------------------
// Kernel 4: r_dn[p, :] = g[p, :] @ lora_down_A[expert(p)]             (R=16)
// ---------------------------------------------------------------------------
__global__ __launch_bounds__(128)
void lora_r_dn(const float* __restrict__ g, const float* __restrict__ A,
               const int* __restrict__ offs, const int* __restrict__ pexp,
               float* __restrict__ rout)
{
    int p = blockIdx.x;
    if (p >= offs[E_]) return;
    int e = pexp[p];
    const float* gr = g + (size_t)p * F_;
    const float* Ae = A + (size_t)e * F_ * R_;
    int r = threadIdx.x >> 3, l = threadIdx.x & 7;
    float acc = 0.f;
    for (int f = l; f < F_; f += 8) acc += gr[f] * Ae[(size_t)f * R_ + r];
    acc += __shfl_xor(acc, 1);
    acc += __shfl_xor(acc, 2);
    acc += __shfl_xor(acc, 4);
    if (l == 0) rout[(size_t)p * R_ + r] = acc;
}

// ---------------------------------------------------------------------------
// Kernel 3: gate_up GEMM + LoRA + bias + clipped-SiLU-GLU -> g workspace.
// Block = 256 threads = 8 waves. Tile: 16 tokens x 128 D-columns (16/wave).
// WMMA f32 16x16x4 over K=H, double-buffered async global->LDS x staging.
// ---------------------------------------------------------------------------
__global__ __launch_bounds__(256)
void moe_gemm1(const float* __restrict__ x, const float* __restrict__ W,
               const float* __restrict__ bias, const float* __restrict__ Blo,
               const float* __restrict__ rgu, const int* __restrict__ offs,
               const int* __restrict__ ptok, float* __restrict__ gbuf)
{
    int e    = blockIdx.z;
    int base = offs[e];
    int cnt  = offs[e + 1] - base;
    int row0 = blockIdx.y * 16;
    if (row0 >= cnt) return;
    int n0 = blockIdx.x * 128;

    __shared__ __align__(16) float xs[2][16][68];  // double-buffered, padded
    __shared__ float rs[16][16];
    __shared__ int   ts[16];

    int tid = threadIdx.x;
    if (tid < 16) {
        int rr = row0 + tid;
        ts[tid] = (rr < cnt) ? ptok[base + rr] : -1;
    }
    rs[tid >> 4][tid & 15] =
        rgu[(size_t)(base + row0 + (tid >> 4)) * R_ + (tid & 15)];
    __syncthreads();

    int wave = tid >> 5, lane = tid & 31;
    int half = lane >> 4, mrow = lane & 15;
    int col  = n0 + wave * 16 + mrow;
    const float* Wc = W + (size_t)e * H_ * D_ + col;   // Wc[h*D_] = W[e,h,col]

    int lr = tid >> 4;            // 0..15 : token row this thread stages
    int lc = (tid & 15) * 4;      // 0,4,...,60
    int tok = ts[lr];
    if (tok < 0) tok = 0;         // padded rows: garbage only hits unstored rows
    const float* src = x + (size_t)tok * H_ + lc;
    unsigned ldsa[2] = { (unsigned)(uintptr_t)&xs[0][lr][lc],
                         (unsigned)(uintptr_t)&xs[1][lr][lc] };

    // Prologue: kick off chunk 0 into buffer 0.
    async_copy_b128(ldsa[0], (uint64_t)(uintptr_t)(src));

    v8f c = {};
    for (int k0 = 0; k0 < H_; k0 += 64) {
        int cur = (k0 >> 6) & 1;
        if (k0 + 64 < H_) __builtin_prefetch(Wc + (size_t)(k0 + 64) * D_, 0, 1);
        __syncthreads();                       // buffer (cur^1) free for overwrite
        if (k0 + 64 < H_) {
            async_copy_b128(ldsa[cur ^ 1], (uint64_t)(uintptr_t)(src + k0 + 64));
            async_wait1();                     // oldest (= cur chunk) complete
        } else {
            async_wait0();
        }
        __syncthreads();                       // cur chunk visible to all waves
        const float (*xb)[68] = xs[cur];
#pragma unroll
        for (int kk = 0; kk < 64; kk += 4) {
            v2f a, b;
            a.x = xb[mrow][kk + 2 * half];
            a.y = xb[mrow][kk + 2 * half + 1];
            const float* wp = Wc + (size_t)(k0 + kk + 2 * half) * D_;
            b.x = wp[0];
            b.y = wp[D_];
            c = __builtin_amdgcn_wmma_f32_16x16x4_f32(
                    false, a, false, b, (short)0, c, false, false);
        }
    }

    // LoRA rank-16 + bias epilogue (SCALING = alpha/r = 1.0 here, kept symbolic)
    float bc[R_];
    const float* Bcol = Blo + (size_t)e * R_ * D_ + col;
#pragma unroll
    for (int r = 0; r < R_; ++r) bc[r] = Bcol[(size_t)r * D_];
    float biasc = bias[(size_t)e * D_ + col];

#pragma unroll
    for (int v = 0; v < 8; ++v) {
        int M = v + 8 * half;                 // output token row in tile
        float val = c[v];
#pragma unroll
        for (int r = 0; r < R_; ++r) val += rs[M][r] * bc[r] * SCALING_;
        val += biasc;
        float other = __shfl_xor(val, 1);     // pair gate(even col)/up(odd col)
        if ((mrow & 1) == 0) {
            float gate = fminf(val, LIMIT_);
            float up   = fminf(fmaxf(other, -LIMIT_), LIMIT_);
            float glu  = gate / (1.f + __expf(-ACTA_ * gate));
            float gv   = (up + 1.f) * glu;
            int rr = row0 + M;
            if (rr < cnt)
                gbuf[(size_t)(base + rr) * F_ + (col >> 1)] = gv;
        }
    }
}

// ---------------------------------------------------------------------------
// Kernel 5: down GEMM + LoRA + bias, scaled by routing weight, atomic combine.
// ---------------------------------------------------------------------------
__global__ __launch_bounds__(256)
void moe_gemm2(const float* __restrict__ gbuf, const float* __restrict__ W,
               const float* __restrict__ bias, const float* __restrict__ Blo,
               const float* __restrict__ rdn, const int* __restrict__ offs,
               const int* __restrict__ ptok, const float* __restrict__ pw,
               float* __restrict__ out)
{
    int e    = blockIdx.z;
    int base = offs[e];
    int cnt  = offs[e + 1] - base;
    int row0 = blockIdx.y * 16;
    if (row0 >= cnt) return;
    int n0 = blockIdx.x * 128;

    __shared__ __align__(16) float xs[2][16][68];
    __shared__ float rs[16][16];
    __shared__ int   ts[16];
    __shared__ float wwt[16];

    int tid = threadIdx.x;
    if (tid < 16) {
        int rr = row0 + tid;
        ts[tid]  = (rr < cnt) ? ptok[base + rr] : 0;
        wwt[tid] = (rr < cnt) ? pw[base + rr] : 0.f;
    }
    rs[tid >> 4][tid & 15] =
        rdn[(size_t)(base + row0 + (tid >> 4)) * R_ + (tid & 15)];
    __syncthreads();

    int wave = tid >> 5, lane = tid & 31;
    int half = lane >> 4, mrow = lane & 15;
    int col  = n0 + wave * 16 + mrow;
    const float* Wc = W + (size_t)e * F_ * H_ + col;   // Wc[f*H_] = W[e,f,col]

    int lr = tid >> 4, lc = (tid & 15) * 4;
    const float* src = gbuf + (size_t)(base + row0 + lr) * F_ + lc;
    unsigned ldsa[2] = { (unsigned)(uintptr_t)&xs[0][lr][lc],
                         (unsigned)(uintptr_t)&xs[1][lr][lc] };

    async_copy_b128(ldsa[0], (uint64_t)(uintptr_t)(src));

    v8f c = {};
    for (int k0 = 0; k0 < F_; k0 += 64) {
        int cur = (k0 >> 6) & 1;
        if (k0 + 64 < F_) __builtin_prefetch(Wc + (size_t)(k0 + 64) * H_, 0, 1);
        __syncthreads();
        if (k0 + 64 < F_) {
            async_copy_b128(ldsa[cur ^ 1], (uint64_t)(uintptr_t)(src + k0 + 64));
            async_wait1();
        } else {
            async_wait0();
        }
        __syncthreads();
        const float (*xb)[68] = xs[cur];
#pragma unroll
        for (int kk = 0; kk < 64; kk += 4) {
            v2f a, b;
            a.x = xb[mrow][kk + 2 * half];
            a.y = xb[mrow][kk + 2 * half + 1];
            const float* wp = Wc + (size_t)(k0 + kk + 2 * half) * H_;
            b.x = wp[0];
            b.y = wp[H_];
            c = __builtin_amdgcn_wmma_f32_16x16x4_f32(
                    false, a, false, b, (short)0, c, false, false);
        }
    }

    float bc[R_];
    const float* Bcol = Blo + (size_t)e * R_ * H_ + col;
#pragma unroll
    for (int r = 0; r < R_; ++r) bc[r] = Bcol[(size_t)r * H_];
    float biasc = bias[(size_t)e * H_ + col];

#pragma unroll
    for (int v = 0; v < 8; ++v) {
        int M = v + 8 * half;
        float val = c[v];
#pragma unroll
        for (int r = 0; r < R_; ++r) val += rs[M][r] * bc[r] * SCALING_;
        val += biasc;
        int rr = row0 + M;
        if (rr < cnt)
            atomicAdd(out + (size_t)ts[M] * H_ + col, val * wwt[M]);
    }
}

// ---------------------------------------------------------------------------
// Host-side launcher.
// ---------------------------------------------------------------------------
extern "C" void kernel_launch(void* const* d_in, const int* in_sizes, int n_in,
                              void* d_out, int out_size, void* d_ws, size_t ws_size,
                              hipStream_t stream)
{
    (void)in_sizes; (void)n_in; (void)out_size; (void)ws_size;

    const float* x    = (const float*)d_in[0];   // [T, H]
    const int*   ridx = (const int*)d_in[1];     // [T, K]
    const float* rw   = (const float*)d_in[2];   // [T, K]
    const float* gup  = (const float*)d_in[3];   // [E, H, D]
    const float* gub  = (const float*)d_in[4];   // [E, D]
    const float* dwn  = (const float*)d_in[5];   // [E, F, H]
    const float* dwb  = (const float*)d_in[6];   // [E, H]
    const float* lguA = (const float*)d_in[7];   // [E, H, R]
    const float* lguB = (const float*)d_in[8];   // [E, R, D]
    const float* ldnA = (const float*)d_in[9];   // [E, F, R]
    const float* ldnB = (const float*)d_in[10];  // [E, R, H]
    float* out = (float*)d_out;

    // Workspace carve-up (16B-aligned sections).
    int*   offs = (int*)d_ws;                    // E_+1 ints, padded to 32
    int*   ptok = offs + 32;                     // CAPP_
    int*   pexp = ptok + CAPP_;                  // CAPP_
    float* pw   = (float*)(pexp + CAPP_);        // CAPP_
    float* rgu  = pw + CAPP_;                    // CAPP_ * R_
    float* rdn  = rgu + (size_t)CAPP_ * R_;      // CAPP_ * R_
    float* gbuf = rdn + (size_t)CAPP_ * R_;      // CAPP_ * F_  (~8.2 MB)

    hipMemsetAsync(d_out, 0, (size_t)T_ * H_ * sizeof(float), stream);

    moe_route<<<1, T_, 0, stream>>>(ridx, rw, offs, ptok, pexp, pw);
    lora_r_gu<<<NPAIR_, 128, 0, stream>>>(x, lguA, offs, ptok, pexp, rgu);
    moe_gemm1<<<dim3(D_ / 128, NPAIR_ / 16, E_), 256, 0, stream>>>(
        x, gup, gub, lguB, rgu, offs, ptok, gbuf);
    lora_r_dn<<<NPAIR_, 128, 0, stream>>>(gbuf, ldnA, offs, pexp, rdn);
    moe_gemm2<<<dim3(H_ / 128, NPAIR_ / 16, E_), 256, 0, stream>>>(
        gbuf, dwn, dwb, ldnB, rdn, offs, ptok, pw, out);
}